// MessagePassing_21320217657821
// MI455X (gfx1250) — compile-verified
//
#include <hip/hip_runtime.h>
#include <stdint.h>

// ---------------- config ----------------
#define THREADS   512                 // 16 wave32s
#define WAVES     (THREADS / 32)
#define RNODES    512                 // nodes privatized per block (128 KB LDS accum)
#define TILE      128                 // edges per wave-tile (32 lanes x int4)

// ---------------- CDNA5 (gfx1250) inline-asm helpers ----------------
// Generic pointers into __shared__ carry the LDS byte offset in their low 32
// bits (flat LDS aperture).

// GLOBAL_LOAD_ASYNC_TO_LDS_B128: per lane, 16B memory -> LDS. Tracked by the
// per-wave ASYNCcnt, which is what makes the barrier-free wave-private
// pipeline possible.
__device__ __forceinline__ void async_copy16_to_lds(void* lds_dst, const void* gsrc) {
    const uint32_t lds_off = (uint32_t)(uintptr_t)lds_dst;
    asm volatile("global_load_async_to_lds_b128 %0, %1, off"
                 :
                 : "v"(lds_off), "v"(gsrc)
                 : "memory");
}
#define S_WAIT_ASYNCCNT(N) asm volatile("s_wait_asynccnt " #N ::: "memory")
#define S_WAIT_DSCNT0()    asm volatile("s_wait_dscnt 0x0" ::: "memory")

// DS_ADD_F32: LDS float atomic add, no return (DScnt). Compiler can't see the
// DS counter for asm ops, so we wait explicitly before the final barrier.
__device__ __forceinline__ void lds_atomic_add_f32(uint32_t lds_byte_off, float v) {
    asm volatile("ds_add_f32 %0, %1" :: "v"(lds_byte_off), "v"(v) : "memory");
}

// ---------------- kernel ----------------
__global__ __launch_bounds__(THREADS)
void mp_segment_sum_kernel(const float* __restrict__ x,
                           const int*   __restrict__ srcRow,
                           const int*   __restrict__ dstRow,
                           float*       __restrict__ out,
                           int E, int N)
{
    __shared__ float    acc[RNODES * 64];        // 128 KB block-private accumulator
    __shared__ int      stage[2][WAVES][TILE];   // 16 KB: per-wave double buffers
    __shared__ uint32_t wq[WAVES][TILE];         //  8 KB: wave-private match queues

    const int tid  = threadIdx.x;
    const int lane = tid & 31;
    const int wave = tid >> 5;
    const int g    = lane >> 4;     // 0/1: 16-lane group within the wave
    const int f4   = lane & 15;     // float4 slot within the 64-feature row

    const int node0 = blockIdx.x * RNODES;
    const int Rb    = min(RNODES, N - node0);
    const unsigned laneMaskLt = (1u << lane) - 1u;

    // Zero accumulators (doubles as output zero-init for edge-less nodes).
    for (int i = tid; i < RNODES * 16; i += THREADS)
        ((float4*)acc)[i] = make_float4(0.f, 0.f, 0.f, 0.f);
    __syncthreads();   // barrier #1: acc zeroed before any cross-wave ds_add_f32

    const int fullTiles  = E / TILE;
    const int totalTiles = (E + TILE - 1) / TILE;

    // Per-wave strided tile stream: t = wave, wave+WAVES, ...
    int t = wave;
    if (t < fullTiles)
        async_copy16_to_lds(&stage[0][wave][lane * 4],
                            dstRow + t * TILE + lane * 4);

    int buf = 0;
    for (; t < totalTiles; t += WAVES) {
        const bool curStaged  = (t < fullTiles);
        const int  tn         = t + WAVES;
        const bool nextStaged = (tn < fullTiles);

        if (nextStaged)   // prefetch this wave's next tile into the other buffer
            async_copy16_to_lds(&stage[buf ^ 1][wave][lane * 4],
                                dstRow + tn * TILE + lane * 4);
        if (curStaged) {
            if (nextStaged) { S_WAIT_ASYNCCNT(1); }   // oldest (current buf) landed
            else            { S_WAIT_ASYNCCNT(0); }
        }

        // ---- scan: 4 edges per lane ----
        const int ebase = t * TILE + lane * 4;
        int dv[4];
        if (curStaged) {
            const int4 d4 = *(const int4*)&stage[buf][wave][lane * 4];
            dv[0] = d4.x; dv[1] = d4.y; dv[2] = d4.z; dv[3] = d4.w;
        } else {          // guarded tail tile straight from global
            dv[0] = (ebase + 0 < E) ? dstRow[ebase + 0] : -1;
            dv[1] = (ebase + 1 < E) ? dstRow[ebase + 1] : -1;
            dv[2] = (ebase + 2 < E) ? dstRow[ebase + 2] : -1;
            dv[3] = (ebase + 3 < E) ? dstRow[ebase + 3] : -1;
        }

        // ---- intra-wave ballot compaction into this wave's queue ----
        int qn = 0;
        #pragma unroll
        for (int k = 0; k < 4; ++k) {
            const unsigned r = (unsigned)(dv[k] - node0);
            const bool     m = r < (unsigned)Rb;
            const unsigned mask = (unsigned)__ballot(m);   // wave32: low 32 bits
            if (m) {
                const int pos = qn + __popc(mask & laneMaskLt);
                wq[wave][pos] = (r << 17) | (uint32_t)srcRow[ebase + k];
            }
            qn += __popc(mask);    // uniform across the wave
        }

        // ---- drain own queue: 16 lanes per matched edge, 2 edges at a time ----
        for (int qi = g; qi < qn; qi += 2) {
            const uint32_t pk  = wq[wave][qi];
            const int      s   = (int)(pk & 0x1FFFFu);   // src node (< 2^17)
            const int      rel = (int)(pk >> 17);        // slot in acc
            // one coalesced 256B gather per edge (16 lanes x b128)
            const float4 v = ((const float4*)(x + (size_t)s * 64))[f4];
            const uint32_t off = (uint32_t)(uintptr_t)&acc[rel * 64 + f4 * 4];
            lds_atomic_add_f32(off + 0,  v.x);
            lds_atomic_add_f32(off + 4,  v.y);
            lds_atomic_add_f32(off + 8,  v.z);
            lds_atomic_add_f32(off + 12, v.w);
        }
        buf ^= 1;
    }

    // All asm ds_add_f32 from each wave must land before anyone reads acc.
    S_WAIT_DSCNT0();
    __syncthreads();   // barrier #2: the only other barrier in the kernel

    // ---- flush: exclusive node ownership -> plain coalesced stores ----
    float* obase = out + (size_t)node0 * 64;
    const int nOut4 = Rb * 16;
    for (int i = tid; i < nOut4; i += THREADS)
        ((float4*)obase)[i] = ((const float4*)acc)[i];
}

// ---------------- launcher ----------------
extern "C" void kernel_launch(void* const* d_in, const int* in_sizes, int n_in,
                              void* d_out, int out_size, void* d_ws, size_t ws_size,
                              hipStream_t stream) {
    const float* x    = (const float*)d_in[0];
    const int*   edge = (const int*)d_in[1];   // edge_index flat: row0=src, row1=dst
    const int    E    = in_sizes[1] / 2;
    const int*   src  = edge;
    const int*   dst  = edge + E;
    float*       out  = (float*)d_out;

    const int N      = out_size / 64;          // 64 features per node
    const int blocks = (N + RNODES - 1) / RNODES;

    mp_segment_sum_kernel<<<blocks, THREADS, 0, stream>>>(x, src, dst, out, E, N);
}